// GIN_60095182405865
// MI455X (gfx1250) — compile-verified
//
#include <hip/hip_runtime.h>
#include <hip/hip_bf16.h>

typedef __attribute__((ext_vector_type(2))) float v2f;
typedef __attribute__((ext_vector_type(8))) float v8f;

#define LDSW 136   // 128 + 8 pad: (2*136) % 64 == 16 -> half-wave B rows hit disjoint bank windows

// ---------------------------------------------------------------------------
// Edge scatter-add: agg[dst] += x[src], one edge per wave32, float4 per lane.
// ---------------------------------------------------------------------------
__global__ void gin_scatter(const int* __restrict__ src, const int* __restrict__ dst,
                            const float* __restrict__ x, float* __restrict__ agg, int nE) {
    const int wave   = (blockIdx.x * blockDim.x + threadIdx.x) >> 5;
    const int lane   = threadIdx.x & 31;
    const int nWaves = (gridDim.x * blockDim.x) >> 5;
    for (int e = wave; e < nE; e += nWaves) {
        const int s = src[e];
        const int d = dst[e];
        const float4 v = *(const float4*)(x + (long)s * 128 + (lane << 2));
        float* b = agg + (long)d * 128 + (lane << 2);
        atomicAdd(b + 0, v.x);
        atomicAdd(b + 1, v.y);
        atomicAdd(b + 2, v.z);
        atomicAdd(b + 3, v.w);
    }
}

// ---------------------------------------------------------------------------
// [M,128] @ [128,128] + bias (+ReLU) via V_WMMA_F32_16X16X4_F32.
// Workgroup = 8 waves; wave w computes the 16x16 tile at cols [16w,16w+16).
// A 16x4 f32 layout: lanes 0-15 hold K=k,k+1 of row M=lane; lanes 16-31 K=k+2,k+3.
// C/D: VGPR j = row j (lanes 0-15) / row j+8 (lanes 16-31), N = lane%16.
// ---------------------------------------------------------------------------
__global__ __launch_bounds__(256) void gin_gemm128(const float* __restrict__ A,
                                                   const float* __restrict__ W,
                                                   const float* __restrict__ bias,
                                                   float* __restrict__ out,
                                                   int doRelu) {
    __shared__ float sW[128 * LDSW];
    const int tid = threadIdx.x;

    // Stage 128x128 weight block into padded LDS (float4 per thread-iter).
    for (int i = tid; i < 128 * 32; i += 256) {
        const int row = i >> 5;
        const int c4  = (i & 31) << 2;
        const float4 v = ((const float4*)W)[i];
        *(float4*)&sW[row * LDSW + c4] = v;
    }
    __syncthreads();

    const int wave = tid >> 5;
    const int lane = tid & 31;
    const int half = lane >> 4;      // 0: lanes 0-15, 1: lanes 16-31
    const int lm   = lane & 15;
    const int n0   = wave << 4;
    const long r0  = (long)blockIdx.x << 4;

    const float* aRow = A + (r0 + lm) * 128 + (half << 1);
    const float* bp   = sW + (half << 1) * LDSW + n0 + lm;

    v8f c = {};
#pragma unroll
    for (int k = 0; k < 128; k += 4) {
        v2f a = *(const v2f*)(aRow + k);           // global_load_b64, rows stay in L2/L0
        v2f b;
        b.x = bp[(k + 0) * LDSW];                  // ds_load_b32, conflict-free windows
        b.y = bp[(k + 1) * LDSW];
        c = __builtin_amdgcn_wmma_f32_16x16x4_f32(
                /*neg_a=*/false, a, /*neg_b=*/false, b,
                /*c_mod=*/(short)0, c, /*reuse_a=*/false, /*reuse_b=*/false);
    }

    const int   n  = n0 + lm;
    const float bn = bias[n];
    float* op = out + (r0 + (half << 3)) * 128 + n;
#pragma unroll
    for (int j = 0; j < 8; ++j) {
        float v = c[j] + bn;
        if (doRelu) v = fmaxf(v, 0.0f);
        op[(long)j * 128] = v;
    }
}

// ---------------------------------------------------------------------------
// BatchNorm statistics: per-column sum / sumsq partials + global atomics.
// ---------------------------------------------------------------------------
__global__ void gin_bnstats(const float* __restrict__ h, float* __restrict__ gsum,
                            float* __restrict__ gsq, int nRows) {
    const int c = threadIdx.x;     // blockDim == 128
    float s = 0.0f, q = 0.0f;
    for (int r = blockIdx.x; r < nRows; r += gridDim.x) {
        const float v = h[(long)r * 128 + c];
        s += v;
        q += v * v;
    }
    atomicAdd(&gsum[c], s);
    atomicAdd(&gsq[c], q);
}

__global__ void gin_bnfinal(const float* __restrict__ gsum, const float* __restrict__ gsq,
                            const float* __restrict__ gamma, const float* __restrict__ beta,
                            float* __restrict__ scale, float* __restrict__ shift, float invN) {
    const int c = threadIdx.x;     // one block of 128
    const float m   = gsum[c] * invN;
    const float var = gsq[c] * invN - m * m;
    const float sc  = gamma[c] * rsqrtf(var + 1e-5f);
    scale[c] = sc;
    shift[c] = beta[c] - m * sc;
}

__global__ void gin_bnapply_relu(const float* __restrict__ h, const float* __restrict__ scale,
                                 const float* __restrict__ shift, float* __restrict__ out, long n) {
    long i = (long)blockIdx.x * blockDim.x + threadIdx.x;
    const long stride = (long)gridDim.x * blockDim.x;
    for (; i < n; i += stride) {
        const int c = (int)(i & 127);
        out[i] = fmaxf(fmaf(h[i], scale[c], shift[c]), 0.0f);
    }
}

__global__ void gin_bnapply_pool(const float* __restrict__ h, const float* __restrict__ scale,
                                 const float* __restrict__ shift, const int* __restrict__ batch,
                                 float* __restrict__ pooled, int nRows) {
    const int c        = threadIdx.x & 127;
    const int rowInBlk = threadIdx.x >> 7;          // blockDim 256 -> 2 rows per block
    const int stride   = gridDim.x << 1;
    for (int r = (blockIdx.x << 1) + rowInBlk; r < nRows; r += stride) {
        const int g = batch[r];
        const float v = fmaf(h[(long)r * 128 + c], scale[c], shift[c]);
        atomicAdd(&pooled[(long)g * 128 + c], v);
    }
}

// ---------------------------------------------------------------------------
// logits = pooled[512,128] @ lw[128,10] + lb   (tiny; VALU dot products)
// ---------------------------------------------------------------------------
__global__ void gin_logits(const float* __restrict__ pooled, const float* __restrict__ lw,
                           const float* __restrict__ lb, float* __restrict__ out, int nG) {
    const int i = blockIdx.x * blockDim.x + threadIdx.x;
    if (i >= nG * 10) return;
    const int g = i / 10;
    const int c = i % 10;
    float acc = lb[c];
    const float* p = pooled + (long)g * 128;
#pragma unroll 4
    for (int k = 0; k < 128; ++k) acc = fmaf(p[k], lw[k * 10 + c], acc);
    out[i] = acc;
}

__global__ void gin_zero(float* __restrict__ p, int n) {
    const int i = blockIdx.x * blockDim.x + threadIdx.x;
    if (i < n) p[i] = 0.0f;
}

// ---------------------------------------------------------------------------
extern "C" void kernel_launch(void* const* d_in, const int* in_sizes, int n_in,
                              void* d_out, int out_size, void* d_ws, size_t ws_size,
                              hipStream_t stream) {
    const float* x    = (const float*)d_in[0];
    const int*   edge = (const int*)  d_in[1];
    const int*   batch= (const int*)  d_in[2];
    const float* w1_0 = (const float*)d_in[3];
    const float* b1_0 = (const float*)d_in[4];
    const float* w2_0 = (const float*)d_in[5];
    const float* b2_0 = (const float*)d_in[6];
    const float* g0   = (const float*)d_in[7];
    const float* be0  = (const float*)d_in[8];
    const float* w1_1 = (const float*)d_in[9];
    const float* b1_1 = (const float*)d_in[10];
    const float* w2_1 = (const float*)d_in[11];
    const float* b2_1 = (const float*)d_in[12];
    const float* g1   = (const float*)d_in[13];
    const float* be1  = (const float*)d_in[14];
    const float* lw   = (const float*)d_in[15];
    const float* lb   = (const float*)d_in[16];

    const int  NN = in_sizes[2];          // 100000 nodes (batch vector length)
    const int  NE = in_sizes[1] / 2;      // 1600000 edges
    const int  NG = 512;
    const long ND = (long)NN * 128;

    float* buf0   = (float*)d_ws;
    float* buf1   = buf0 + ND;
    float* buf2   = buf1 + ND;
    float* gsum   = buf2 + ND;
    float* gsq    = gsum + 128;
    float* scale  = gsq  + 128;
    float* shift  = scale + 128;
    float* pooled = shift + 128;

    const int* src  = edge;
    const int* dstI = edge + NE;
    const dim3 b256(256);
    const int  mTiles = NN / 16;          // 6250 exact

    // ---- layer 0: GINConv -> BN -> ReLU ----
    hipMemcpyAsync(buf0, x, ND * sizeof(float), hipMemcpyDeviceToDevice, stream);
    gin_scatter<<<1024, b256, 0, stream>>>(src, dstI, x, buf0, NE);
    gin_gemm128<<<mTiles, b256, 0, stream>>>(buf0, w1_0, b1_0, buf1, 1);
    gin_gemm128<<<mTiles, b256, 0, stream>>>(buf1, w2_0, b2_0, buf2, 0);
    gin_zero<<<1, 256, 0, stream>>>(gsum, 256);                 // gsum+gsq contiguous
    gin_bnstats<<<512, 128, 0, stream>>>(buf2, gsum, gsq, NN);
    gin_bnfinal<<<1, 128, 0, stream>>>(gsum, gsq, g0, be0, scale, shift, 1.0f / (float)NN);
    gin_bnapply_relu<<<2048, b256, 0, stream>>>(buf2, scale, shift, buf0, ND);

    // ---- layer 1: GINConv -> BN -> pool ----
    hipMemcpyAsync(buf1, buf0, ND * sizeof(float), hipMemcpyDeviceToDevice, stream);
    gin_scatter<<<1024, b256, 0, stream>>>(src, dstI, buf0, buf1, NE);
    gin_gemm128<<<mTiles, b256, 0, stream>>>(buf1, w1_1, b1_1, buf2, 1);
    gin_gemm128<<<mTiles, b256, 0, stream>>>(buf2, w2_1, b2_1, buf1, 0);
    gin_zero<<<1, 256, 0, stream>>>(gsum, 256);
    gin_bnstats<<<512, 128, 0, stream>>>(buf1, gsum, gsq, NN);
    gin_bnfinal<<<1, 128, 0, stream>>>(gsum, gsq, g1, be1, scale, shift, 1.0f / (float)NN);
    gin_zero<<<(NG * 128 + 255) / 256, 256, 0, stream>>>(pooled, NG * 128);
    gin_bnapply_pool<<<2048, b256, 0, stream>>>(buf1, scale, shift, batch, pooled, NN);

    // ---- classifier ----
    gin_logits<<<(NG * 10 + 255) / 256, 256, 0, stream>>>(pooled, lw, lb, (float*)d_out, NG);
}